// GCN_22325240005451
// MI455X (gfx1250) — compile-verified
//
#include <hip/hip_runtime.h>
#include <hip/hip_bf16.h>

typedef float v2f __attribute__((ext_vector_type(2)));
typedef float v8f __attribute__((ext_vector_type(8)));

#define HDIM 96      // hidden feature dim = N of both GEMMs
#define NTILE 6      // 96 / 16
#define KMAX 128     // max K (layer-1 GEMM)

// ---------------- small elementwise kernels ----------------

__global__ void fill_zero_k(float* __restrict__ p, int n) {
  int i = blockIdx.x * blockDim.x + threadIdx.x;
  if (i < n) p[i] = 0.0f;
}

__global__ void deg_count_k(const int* __restrict__ dst, float* __restrict__ deg, int E) {
  int i = blockIdx.x * blockDim.x + threadIdx.x;
  if (i < E) atomicAdd(&deg[dst[i]], 1.0f);
}

__global__ void make_dinv_k(const float* __restrict__ deg, float* __restrict__ dinv, int n) {
  int i = blockIdx.x * blockDim.x + threadIdx.x;
  if (i < n) dinv[i] = rsqrtf(deg[i] + 1.0f);   // deg includes self-loop
}

// AGG[i,f] = H[i,f] * dinv[i]^2 + bias[f]   (self-loop term + bias)
__global__ void agg_init_k(const float* __restrict__ H, const float* __restrict__ dinv,
                           const float* __restrict__ bias, float* __restrict__ AGG, int n) {
  int i = blockIdx.x * blockDim.x + threadIdx.x;
  int total = n * HDIM;
  if (i < total) {
    int node = i / HDIM;
    int f    = i - node * HDIM;
    float di = dinv[node];
    AGG[i] = H[i] * di * di + bias[f];
  }
}

// one wave per edge: AGG[dst] += H[src] * dinv[src]*dinv[dst]
__global__ void scatter_edges_k(const int* __restrict__ src, const int* __restrict__ dst,
                                const float* __restrict__ H, const float* __restrict__ dinv,
                                float* __restrict__ AGG, int E) {
  int gid  = blockIdx.x * blockDim.x + threadIdx.x;
  int e    = gid >> 5;
  int lane = gid & 31;
  if (e >= E) return;
  int s = src[e];
  int d = dst[e];
  float norm = dinv[s] * dinv[d];
  const float* hs = H   + (size_t)s * HDIM;
  float*       ad = AGG + (size_t)d * HDIM;
  #pragma unroll
  for (int f = lane; f < HDIM; f += 32)
    atomicAdd(&ad[f], hs[f] * norm);
}

__global__ void relu_ip_k(float* __restrict__ p, int n) {
  int i = blockIdx.x * blockDim.x + threadIdx.x;
  if (i < n) p[i] = fmaxf(p[i], 0.0f);
}

// ---------------- fp32 WMMA GEMM:  C[M x 96] = A[M x K] * B[K x 96] ----------------
// One wave computes a 16-row x 96-col panel: 6 v8f accumulators, K stepped by 4.
// B staged in LDS K-pair-interleaved:  Bs2[kk*96 + col] = { B[2kk][col], B[2kk+1][col] }
// so each WMMA B operand is one naturally-aligned ds_load_b64 (no repack moves).
__global__ __launch_bounds__(256)
void gemm_wmma_f32_k(const float* __restrict__ A, const float* __restrict__ B,
                     float* __restrict__ C, int M, int K) {
  __shared__ v2f Bs2[(KMAX / 2) * HDIM];

  int tid   = threadIdx.x;
  int pairs = (K >> 1) * HDIM;
  for (int i = tid; i < pairs; i += blockDim.x) {
    int kk  = i / HDIM;
    int col = i - kk * HDIM;
    v2f t;
    t.x = B[(2 * kk)     * HDIM + col];
    t.y = B[(2 * kk + 1) * HDIM + col];
    Bs2[i] = t;
  }
  __syncthreads();

  int mt = blockIdx.x * (blockDim.x >> 5) + (tid >> 5);   // 16-row tile index
  if (mt * 16 >= M) return;                               // wave-uniform exit

  int lane = tid & 31;
  int l16  = lane & 15;
  int hi   = lane >> 4;                                   // 0: lanes 0-15, 1: lanes 16-31

  int arow = mt * 16 + l16;
  if (arow >= M) arow = M - 1;                            // clamp (M=50000 is 16-aligned anyway)
  const float* Ap = A + (size_t)arow * K;

  v8f acc[NTILE];
  #pragma unroll
  for (int t = 0; t < NTILE; ++t)
    #pragma unroll
    for (int j = 0; j < 8; ++j) acc[t][j] = 0.0f;

  for (int k = 0; k < K; k += 4) {
    // A 16x4 layout: lane pair holds K = {k+2*hi, k+1+2*hi} -> one 8B global load
    v2f a;
    a.x = Ap[k + 2 * hi];
    a.y = Ap[k + 2 * hi + 1];
    int kk = (k + 2 * hi) >> 1;                           // even row index / 2
    const v2f* brow = &Bs2[kk * HDIM];
    #pragma unroll
    for (int t = 0; t < NTILE; ++t) {
      v2f b = brow[t * 16 + l16];                         // single ds_load_b64
      acc[t] = __builtin_amdgcn_wmma_f32_16x16x4_f32(
          /*neg_a=*/false, a, /*neg_b=*/false, b,
          /*c_mod=*/(short)0, acc[t], /*reuse_a=*/false, /*reuse_b=*/false);
    }
  }

  // C/D layout: VGPR r -> row = r + 8*hi, col = l16 (per n-tile)
  #pragma unroll
  for (int t = 0; t < NTILE; ++t) {
    int col = t * 16 + l16;
    #pragma unroll
    for (int r = 0; r < 8; ++r) {
      int row = mt * 16 + r + 8 * hi;
      if (row < M) C[(size_t)row * HDIM + col] = acc[t][r];
    }
  }
}

// ---------------- classifier: logits = h @ Wc + bc, then log_softmax ----------------
__global__ void classify_k(const float* __restrict__ H, const float* __restrict__ Wc,
                           const float* __restrict__ bc, float* __restrict__ out, int n) {
  int gid  = blockIdx.x * blockDim.x + threadIdx.x;
  int node = gid >> 5;
  int lane = gid & 31;
  if (node >= n) return;
  const float* h = H + (size_t)node * HDIM;
  float l0 = 0.0f, l1 = 0.0f;
  for (int f = lane; f < HDIM; f += 32) {
    float v = h[f];
    l0 += v * Wc[f * 2 + 0];
    l1 += v * Wc[f * 2 + 1];
  }
  #pragma unroll
  for (int off = 16; off > 0; off >>= 1) {
    l0 += __shfl_xor(l0, off, 32);
    l1 += __shfl_xor(l1, off, 32);
  }
  if (lane == 0) {
    l0 += bc[0];
    l1 += bc[1];
    float m   = fmaxf(l0, l1);
    float lse = m + logf(expf(l0 - m) + expf(l1 - m));
    out[(size_t)node * 2 + 0] = l0 - lse;
    out[(size_t)node * 2 + 1] = l1 - lse;
  }
}

// ---------------- driver ----------------
extern "C" void kernel_launch(void* const* d_in, const int* in_sizes, int n_in,
                              void* d_out, int out_size, void* d_ws, size_t ws_size,
                              hipStream_t stream) {
  const float* x  = (const float*)d_in[0];   // [n,128]
  const int*   ei = (const int*)d_in[1];     // [2,E]
  const float* W1 = (const float*)d_in[2];   // [128,96]
  const float* b1 = (const float*)d_in[3];   // [96]
  const float* W2 = (const float*)d_in[4];   // [96,96]
  const float* b2 = (const float*)d_in[5];   // [96]
  const float* Wc = (const float*)d_in[6];   // [96,2]
  const float* bc = (const float*)d_in[7];   // [2]
  float* out = (float*)d_out;                // [n,2]

  const int n = in_sizes[0] / 128;           // 50000
  const int E = in_sizes[1] / 2;             // 800000
  const int* src = ei;
  const int* dst = ei + E;

  float* ws   = (float*)d_ws;
  float* deg  = ws;                          // n
  float* dinv = ws + n;                      // n
  float* Hbuf = ws + 2 * (size_t)n;          // n*96
  float* Abuf = Hbuf + (size_t)n * HDIM;     // n*96

  const int nf      = n * HDIM;
  const dim3 blk(256);
  const int gN  = (n  + 255) / 256;
  const int gE  = (E  + 255) / 256;
  const int gNF = (nf + 255) / 256;
  const int gEdgeWave = (int)(((size_t)E * 32 + 255) / 256);
  const int gNodeWave = (int)(((size_t)n * 32 + 255) / 256);
  const int mtiles    = (n + 15) / 16;       // 3125
  const int gGemm     = (mtiles + 7) / 8;    // 8 waves per 256-thread block

  // normalization coefficients
  fill_zero_k<<<gN, blk, 0, stream>>>(deg, n);
  deg_count_k<<<gE, blk, 0, stream>>>(dst, deg, E);
  make_dinv_k<<<gN, blk, 0, stream>>>(deg, dinv, n);

  // ---- layer 1 ----
  gemm_wmma_f32_k<<<gGemm, blk, 0, stream>>>(x, W1, Hbuf, n, 128);
  agg_init_k<<<gNF, blk, 0, stream>>>(Hbuf, dinv, b1, Abuf, n);
  scatter_edges_k<<<gEdgeWave, blk, 0, stream>>>(src, dst, Hbuf, dinv, Abuf, E);
  relu_ip_k<<<gNF, blk, 0, stream>>>(Abuf, nf);

  // ---- layer 2 ----
  gemm_wmma_f32_k<<<gGemm, blk, 0, stream>>>(Abuf, W2, Hbuf, n, HDIM);
  agg_init_k<<<gNF, blk, 0, stream>>>(Hbuf, dinv, b2, Abuf, n);
  scatter_edges_k<<<gEdgeWave, blk, 0, stream>>>(src, dst, Hbuf, dinv, Abuf, E);
  relu_ip_k<<<gNF, blk, 0, stream>>>(Abuf, nf);

  // ---- classifier + log_softmax ----
  classify_k<<<gNodeWave, blk, 0, stream>>>(Abuf, Wc, bc, out, n);
}